// GraphConvLayer_10995116277856
// MI455X (gfx1250) — compile-verified
//
#include <hip/hip_runtime.h>
#include <hip/hip_bf16.h>

typedef float v2f __attribute__((ext_vector_type(2)));
typedef float v8f __attribute__((ext_vector_type(8)));

#define DIMS 128

// Force the native GLOBAL_ATOMIC_ADD_F32 path (no CAS expansion).
__device__ __forceinline__ void atomic_add_f32(float* addr, float v) {
    unsafeAtomicAdd(addr, v);
}

// ---------------- degree accumulation ----------------
__global__ __launch_bounds__(256) void degree_kernel(const int* __restrict__ src,
                                                     const int* __restrict__ dst,
                                                     float* __restrict__ deg_out,
                                                     float* __restrict__ deg_in, int E) {
    int e = blockIdx.x * 256 + threadIdx.x;
    if (e >= E) return;
    atomic_add_f32(&deg_out[src[e]], 1.0f);
    atomic_add_f32(&deg_in[dst[e]], 1.0f);
}

// ---------------- deg -> 1/sqrt(max(deg,1)) in place ----------------
__global__ __launch_bounds__(256) void rsqrt_kernel(float* __restrict__ deg_out,
                                                    float* __restrict__ deg_in, int N) {
    int i = blockIdx.x * 256 + threadIdx.x;
    if (i >= N) return;
    deg_out[i] = rsqrtf(fmaxf(deg_out[i], 1.0f));
    deg_in[i]  = rsqrtf(fmaxf(deg_in[i], 1.0f));
}

// ---------------- SpMM scatter: agg[dst] += feat[src] * inv_out[src] ----------------
// 32 lanes per edge, float4 per lane (128 floats per edge)
__global__ __launch_bounds__(256) void scatter_kernel(const float* __restrict__ feat,
                                                      const int* __restrict__ src,
                                                      const int* __restrict__ dst,
                                                      const float* __restrict__ inv_out,
                                                      float* __restrict__ agg, int E) {
    long tid = (long)blockIdx.x * 256 + threadIdx.x;
    long e = tid >> 5;
    if (e >= E) return;
    int q = ((int)tid & 31) * 4;
    int s = src[e], t = dst[e];
    float w = inv_out[s];
    const float4 v = *(const float4*)&feat[(long)s * DIMS + q];
    float* base = &agg[(long)t * DIMS + q];
    atomic_add_f32(base + 0, v.x * w);
    atomic_add_f32(base + 1, v.y * w);
    atomic_add_f32(base + 2, v.z * w);
    atomic_add_f32(base + 3, v.w * w);
}

// ---------------- GEMM (fp32 WMMA) + bias + ReLU + LayerNorm ----------------
// One block = 16 output rows x 128 cols. 8 waves, each owns a 16x16 tile.
// x = relu((agg * inv_in[:,None]) @ W + b); out = LN(x)*gamma + beta
__global__ __launch_bounds__(256) void gemm_ln_kernel(const float* __restrict__ agg,
                                                      const float* __restrict__ inv_in,
                                                      const float* __restrict__ W,
                                                      const float* __restrict__ bias,
                                                      const float* __restrict__ gamma,
                                                      const float* __restrict__ beta,
                                                      float* __restrict__ out, int N) {
    __shared__ float xs[16][DIMS];   // A tile (scaled), later reused for x tile
    __shared__ float mu_s[16], rs_s[16];

    const int tid  = threadIdx.x;
    const int wave = tid >> 5;       // 0..7 -> 16-col tile
    const int lane = tid & 31;
    const int row0 = blockIdx.x * 16;

    // Stage scaled A tile into LDS (clamp rows for the tail block)
    for (int i = tid; i < 16 * DIMS; i += 256) {
        int r = i >> 7, c = i & 127;
        int rr = row0 + r; rr = rr < N ? rr : N - 1;
        xs[r][c] = agg[(long)rr * DIMS + c] * inv_in[rr];
    }
    __syncthreads();

    const int n0   = wave * 16;
    const int mrow = lane & 15;             // A row this lane supplies
    const int koff = (lane >> 4) * 2;       // K sub-offset per ISA layout
    const int ncol = n0 + (lane & 15);      // B/D column this lane supplies

    v8f c = {};
#pragma unroll
    for (int k = 0; k < DIMS; k += 4) {
        v2f a = *(const v2f*)&xs[mrow][k + koff];            // ds_load_b64
        v2f b;
        b.x = W[(k + koff) * DIMS + ncol];
        b.y = W[(k + koff + 1) * DIMS + ncol];
        // D = A(16x4) x B(4x16) + C, fp32 exact
        c = __builtin_amdgcn_wmma_f32_16x16x4_f32(false, a, false, b,
                                                  (short)0, c, false, false);
    }
    __syncthreads();   // all waves done reading A tile from LDS

    // bias + relu, deposit x tile back into LDS
    const float bn = bias[ncol];
#pragma unroll
    for (int r = 0; r < 8; ++r) {
        int m = r + 8 * (lane >> 4);        // C/D layout: M = r + 8*(lane>>4)
        float v = c[r] + bn;
        xs[m][ncol] = v > 0.0f ? v : 0.0f;
    }
    __syncthreads();

    // Per-row LayerNorm statistics: 16 threads per row, shfl_xor tree reduce.
    {
        const int row = tid >> 4;       // 0..15
        const int sub = tid & 15;       // 16 threads share one row
        const float4 v0 = *(const float4*)&xs[row][sub * 8 + 0];
        const float4 v1 = *(const float4*)&xs[row][sub * 8 + 4];
        float s  = v0.x + v0.y + v0.z + v0.w + v1.x + v1.y + v1.z + v1.w;
        float ss = v0.x*v0.x + v0.y*v0.y + v0.z*v0.z + v0.w*v0.w
                 + v1.x*v1.x + v1.y*v1.y + v1.z*v1.z + v1.w*v1.w;
#pragma unroll
        for (int m = 1; m < 16; m <<= 1) {   // stays within each 16-lane half-wave
            s  += __shfl_xor(s,  m, 32);
            ss += __shfl_xor(ss, m, 32);
        }
        if (sub == 0) {
            float mu  = s * (1.0f / DIMS);
            float var = ss * (1.0f / DIMS) - mu * mu;
            mu_s[row] = mu;
            rs_s[row] = rsqrtf(var + 1e-5f);
        }
    }
    __syncthreads();

    // Normalize + affine + store
    for (int i = tid; i < 16 * DIMS; i += 256) {
        int r = i >> 7, col = i & 127;
        int rr = row0 + r;
        if (rr < N) {
            float v = (xs[r][col] - mu_s[r]) * rs_s[r] * gamma[col] + beta[col];
            out[(long)rr * DIMS + col] = v;
        }
    }
}

extern "C" void kernel_launch(void* const* d_in, const int* in_sizes, int n_in,
                              void* d_out, int out_size, void* d_ws, size_t ws_size,
                              hipStream_t stream) {
    const float* feat  = (const float*)d_in[0];
    const int*   edges = (const int*)d_in[1];
    const float* W     = (const float*)d_in[2];
    const float* bias  = (const float*)d_in[3];
    const float* gamma = (const float*)d_in[4];
    const float* beta  = (const float*)d_in[5];

    const int N = in_sizes[0] / DIMS;
    const int E = in_sizes[1] / 2;
    const int* src = edges;
    const int* dst = edges + E;

    float* ws      = (float*)d_ws;
    float* deg_out = ws;            // N floats -> becomes inv_sqrt_out
    float* deg_in  = ws + N;        // N floats -> becomes inv_sqrt_in
    float* agg     = ws + 2 * N;    // N*128 floats

    // zero degrees + agg (capture-legal async memset)
    hipMemsetAsync(d_ws, 0, (size_t)(2 * N + (size_t)N * DIMS) * sizeof(float), stream);

    degree_kernel<<<(E + 255) / 256, 256, 0, stream>>>(src, dst, deg_out, deg_in, E);
    rsqrt_kernel<<<(N + 255) / 256, 256, 0, stream>>>(deg_out, deg_in, N);

    long sc_threads = (long)E * 32;
    scatter_kernel<<<(int)((sc_threads + 255) / 256), 256, 0, stream>>>(
        feat, src, dst, deg_out, agg, E);

    gemm_ln_kernel<<<(N + 15) / 16, 256, 0, stream>>>(
        agg, deg_in, W, bias, gamma, beta, (float*)d_out, N);
}